// LSHAttention_12713103196748
// MI455X (gfx1250) — compile-verified
//
#include <hip/hip_runtime.h>
#include <hip/hip_bf16.h>
#include <math.h>
#include <float.h>
#include <stdint.h>

typedef __attribute__((ext_vector_type(2))) float v2f;
typedef __attribute__((ext_vector_type(8))) float v8f;

#define B_  16
#define S_  4096
#define D_  64
#define H_  8
#define NB_ 64          // n_buckets
#define M_  64          // bucket_size
#define C_  512         // chunks per batch = H_*S_/M_
#define HS_ (H_*S_)     // 32768

// ---------------------------------------------------------------------------
// K1: LSH hashing. One block = 64 tokens of one (b,h).
// rotated = qk_row @ rot[:,h,:] (64x32), argmax over [r, -r] -> bucket.
// ---------------------------------------------------------------------------
__global__ __launch_bounds__(256) void k_hash(const float* __restrict__ qk,
                                              const float* __restrict__ rot,
                                              unsigned* __restrict__ bucket) {
    __shared__ float rotl[D_ * 32];
    __shared__ float qkt[64 * 68];
    __shared__ float redv[256];
    __shared__ int   redi[256];
    int tid = threadIdx.x;
    int blk = blockIdx.x;
    int s0 = (blk & 63) * 64;
    int h  = (blk >> 6) & 7;
    int b  = blk >> 9;
    // rotations layout (1, D, H, 32): element (f,h,i) at (f*H + h)*32 + i
    for (int lin = tid; lin < D_ * 32; lin += 256) {
        int f = lin >> 5, i = lin & 31;
        rotl[lin] = rot[(f * H_ + h) * 32 + i];
    }
    for (int e = 0; e < 16; ++e) {
        int lin = e * 256 + tid;
        int r = lin >> 6, c = lin & 63;
        qkt[r * 68 + c] = qk[((size_t)(b * S_ + s0 + r)) * D_ + c];
    }
    __syncthreads();
    int tok = tid >> 2, part = tid & 3, i0 = part * 8;
    float acc[8];
#pragma unroll
    for (int j = 0; j < 8; ++j) acc[j] = 0.f;
    for (int f = 0; f < D_; ++f) {
        float qv = qkt[tok * 68 + f];
#pragma unroll
        for (int j = 0; j < 8; ++j) acc[j] = fmaf(qv, rotl[f * 32 + i0 + j], acc[j]);
    }
    float bv = -FLT_MAX; int bi = 0;
#pragma unroll
    for (int j = 0; j < 8; ++j) { if (acc[j]  > bv) { bv = acc[j];  bi = i0 + j; } }
#pragma unroll
    for (int j = 0; j < 8; ++j) { if (-acc[j] > bv) { bv = -acc[j]; bi = 32 + i0 + j; } }
    redv[tid] = bv; redi[tid] = bi;
    __syncthreads();
    if (part == 0) {
        float v0 = redv[tid]; int ii = redi[tid];
#pragma unroll
        for (int p = 1; p < 4; ++p) {
            float vv = redv[tid + p];
            if (vv > v0) { v0 = vv; ii = redi[tid + p]; }
        }
        bucket[((size_t)(b * H_ + h)) * S_ + s0 + tok] = (unsigned)(h * NB_ + ii);
    }
}

// ---------------------------------------------------------------------------
// K2: stable counting sort of 4096 tokens into 64 buckets, per (b,h).
// Produces sorted_s (original positions in sorted order) and inv (undo_sort).
// ---------------------------------------------------------------------------
__global__ __launch_bounds__(256) void k_sort(const unsigned* __restrict__ bucket,
                                              unsigned* __restrict__ sorted_s,
                                              unsigned* __restrict__ inv) {
    extern __shared__ unsigned smem2[];
    unsigned* cnt  = smem2;             // 64*256
    unsigned* part = smem2 + 64 * 256;  // 256
    int tid = threadIdx.x;
    int blk = blockIdx.x;
    int b = blk >> 3, h = blk & 7;
    const unsigned* bkp = bucket + ((size_t)(b * H_ + h)) * S_;

    for (int i = tid; i < 64 * 256; i += 256) cnt[i] = 0;
    __syncthreads();
    unsigned ub[16];
#pragma unroll
    for (int k = 0; k < 16; ++k) ub[k] = bkp[tid * 16 + k] - (unsigned)(h * NB_);
#pragma unroll
    for (int k = 0; k < 16; ++k) cnt[ub[k] * 256 + tid]++;  // cell owned by tid
    __syncthreads();
    // flat exclusive scan over 16384 cells ordered (bucket, thread)
    unsigned ssum = 0;
    for (int k = 0; k < 64; ++k) ssum += cnt[tid * 64 + k];
    part[tid] = ssum;
    __syncthreads();
    if (tid == 0) {
        unsigned run0 = 0;
        for (int j = 0; j < 256; ++j) { unsigned t = part[j]; part[j] = run0; run0 += t; }
    }
    __syncthreads();
    unsigned run = part[tid];
    for (int k = 0; k < 64; ++k) { unsigned t = cnt[tid * 64 + k]; cnt[tid * 64 + k] = run; run += t; }
    __syncthreads();
    size_t baseOut = (size_t)b * HS_ + (size_t)h * S_;
#pragma unroll
    for (int k = 0; k < 16; ++k) {
        int s = tid * 16 + k;
        unsigned bu = ub[k];
        unsigned off = cnt[bu * 256 + tid];
        cnt[bu * 256 + tid] = off + 1;
        sorted_s[baseOut + off] = (unsigned)s;
        inv[baseOut + s] = (unsigned)(h * S_ + off);
    }
}

// ---------------------------------------------------------------------------
// K3: chunk attention with WMMA f32 16x16x4. One block = one (b, chunk).
// Gathers rows via async global->LDS (ASYNCcnt), pre-normalizes bk,
// dots = bq @ bk^T * 0.125, masks, softmax, bo = attn @ bv.
// ---------------------------------------------------------------------------
__global__ __launch_bounds__(256) void k_attn(const float* __restrict__ qk,
                                              const float* __restrict__ v,
                                              const unsigned* __restrict__ sorted_s,
                                              float* __restrict__ so,
                                              float* __restrict__ slog) {
    extern __shared__ char smem[];
    float* raw  = (float*)smem;            // 128 x 68 (raw qk rows: cur chunk, prev chunk)
    float* bkN  = raw + 128 * 68;          // 128 x 68 (L2-normalized k rows)
    float* bvL  = bkN + 128 * 68;          // 128 x 68 (v rows)
    float* attn = bvL + 128 * 68;          // 64 x 132 (dots -> attn probs)
    int*   tval = (int*)(attn + 64 * 132); // 128 token positions
    float* scl  = (float*)(tval + 128);    // 128 row norm reciprocals

    int tid = threadIdx.x;
    int blk = blockIdx.x;
    int b = blk >> 9, c = blk & 511;
    int cprev = (c + C_ - 1) & (C_ - 1);
    const unsigned* ss = sorted_s + (size_t)b * HS_;
    if (tid < 128) {
        int idx = (tid < 64) ? (c * M_ + tid) : (cprev * M_ + tid - 64);
        tval[tid] = (int)ss[idx];
    }
    __syncthreads();
    // Async gather: 512 rows x 64B each for qk and v (16B per lane-op).
    {
        uint64_t qbase = (uint64_t)(uintptr_t)qk;
        uint64_t vbase = (uint64_t)(uintptr_t)v;
#pragma unroll
        for (int e = 0; e < 8; ++e) {
            int lin = e * 256 + tid;           // 0..2047
            int r = lin >> 2, seg = lin & 3;
            unsigned goff = (((unsigned)(b * S_ + tval[r]) << 6) + ((unsigned)seg << 2)) << 2;
            unsigned l1 = (unsigned)(uintptr_t)(raw + r * 68 + seg * 4);
            unsigned l2 = (unsigned)(uintptr_t)(bvL + r * 68 + seg * 4);
            asm volatile("global_load_async_to_lds_b128 %0, %1, %2"
                         :: "v"(l1), "v"(goff), "s"(qbase) : "memory");
            asm volatile("global_load_async_to_lds_b128 %0, %1, %2"
                         :: "v"(l2), "v"(goff), "s"(vbase) : "memory");
        }
        asm volatile("s_wait_asynccnt 0" ::: "memory");
    }
    __syncthreads();
    // per-row L2 norm reciprocal
    if (tid < 128) {
        float sum = 0.f;
        const float* rr = raw + tid * 68;
#pragma unroll
        for (int j = 0; j < 64; ++j) sum = fmaf(rr[j], rr[j], sum);
        scl[tid] = 1.0f / fmaxf(sqrtf(sum), 1e-12f);
    }
    __syncthreads();
    // pre-normalize bk so the WMMA inner loop is pure ds_load -> wmma
    for (int e = 0; e < 32; ++e) {
        int lin = e * 256 + tid;
        int r = lin >> 6, col = lin & 63;
        bkN[r * 68 + col] = raw[r * 68 + col] * scl[r];
    }
    __syncthreads();

    int lane = tid & 31, wave = tid >> 5;
    int mt = wave >> 1;                 // M tile 0..3
    int nh = (wave & 1) * 4;            // N tile base 0 or 4 (GEMM1)
    int lrow = lane & 15;
    int ksel = (lane >> 4) << 1;        // lanes 16-31 hold K+2
    int arow = mt * 16 + lrow;

    // ---- GEMM1: dots(64x128) = bq(raw rows 0..63) @ bkN^T ----
    v8f cacc[4];
#pragma unroll
    for (int n = 0; n < 4; ++n)
#pragma unroll
        for (int j = 0; j < 8; ++j) cacc[n][j] = 0.f;
    for (int kt = 0; kt < 16; ++kt) {
        int kk = kt * 4 + ksel;
        v2f a;
        a.x = raw[arow * 68 + kk];
        a.y = raw[arow * 68 + kk + 1];
#pragma unroll
        for (int n = 0; n < 4; ++n) {
            int brow = (nh + n) * 16 + lrow;   // bk row (0..127)
            v2f bf;
            bf.x = bkN[brow * 68 + kk];
            bf.y = bkN[brow * 68 + kk + 1];
            cacc[n] = __builtin_amdgcn_wmma_f32_16x16x4_f32(
                false, a, false, bf, (short)0, cacc[n], false, false);
        }
    }
    // spill C to LDS with scale + masks
    {
        int rbase = mt * 16 + ((lane >> 4) << 3);
#pragma unroll
        for (int n = 0; n < 4; ++n) {
            int col = (nh + n) * 16 + lrow;
            int tk = tval[col];
#pragma unroll
            for (int vv = 0; vv < 8; ++vv) {
                int r = rbase + vv;
                float val = cacc[n][vv] * 0.125f;   // D^-0.5
                int tq = tval[r];
                if (tq < tk)  val = -FLT_MAX;       // causal
                if (tq == tk) val = -50000.0f;      // self-attn
                attn[r * 132 + col] = val;
            }
        }
    }
    __syncthreads();
    // ---- softmax + logsumexp, one row per thread ----
    if (tid < 64) {
        float* rowp = attn + tid * 132;
        float mx = -FLT_MAX;
        for (int j = 0; j < 128; ++j) mx = fmaxf(mx, rowp[j]);
        float sum = 0.f;
        for (int j = 0; j < 128; ++j) {
            float e = expf(rowp[j] - mx);
            rowp[j] = e;
            sum += e;
        }
        float rs = 1.0f / sum;
        for (int j = 0; j < 128; ++j) rowp[j] *= rs;
        slog[(size_t)b * HS_ + c * M_ + tid] = mx + logf(sum);
    }
    __syncthreads();
    // ---- GEMM2: bo(64x64) = attn(64x128) @ bv(128x64) ----
    v8f oacc[2];
#pragma unroll
    for (int n = 0; n < 2; ++n)
#pragma unroll
        for (int j = 0; j < 8; ++j) oacc[n][j] = 0.f;
    int nb2 = (wave & 1) * 2;
    for (int kt = 0; kt < 32; ++kt) {
        int kk = kt * 4 + ksel;
        v2f a;
        a.x = attn[arow * 132 + kk];
        a.y = attn[arow * 132 + kk + 1];
#pragma unroll
        for (int n = 0; n < 2; ++n) {
            v2f bf;
            bf.x = bvL[kk * 68 + (nb2 + n) * 16 + lrow];
            bf.y = bvL[(kk + 1) * 68 + (nb2 + n) * 16 + lrow];
            oacc[n] = __builtin_amdgcn_wmma_f32_16x16x4_f32(
                false, a, false, bf, (short)0, oacc[n], false, false);
        }
    }
    {
        int rbase = mt * 16 + ((lane >> 4) << 3);
        size_t obase = ((size_t)b * HS_ + (size_t)c * M_) * D_;
#pragma unroll
        for (int n = 0; n < 2; ++n) {
            int col = (nb2 + n) * 16 + lrow;
#pragma unroll
            for (int vv = 0; vv < 8; ++vv) {
                int r = rbase + vv;
                so[obase + (size_t)r * D_ + col] = oacc[n][vv];
            }
        }
    }
}

// ---------------------------------------------------------------------------
// K4: unsort + combine hash rounds via logsumexp reweighting.
// One 256-thread block = 4 tokens x 64 dims.
// ---------------------------------------------------------------------------
__global__ __launch_bounds__(256) void k_comb(const float* __restrict__ so,
                                              const float* __restrict__ slog,
                                              const unsigned* __restrict__ inv,
                                              float* __restrict__ out) {
    int tid = threadIdx.x;
    int gid = blockIdx.x * 4 + (tid >> 6);
    int d = tid & 63;
    int b = gid >> 12;
    int s = gid & 4095;
    const unsigned* ip = inv + (size_t)b * HS_;
    const float* sl = slog + (size_t)b * HS_;
    unsigned pos[H_];
    float lg[H_];
    float mx = -FLT_MAX;
#pragma unroll
    for (int h = 0; h < H_; ++h) {
        unsigned p = ip[h * S_ + s];
        pos[h] = p;
        float l = sl[p];
        lg[h] = l;
        mx = fmaxf(mx, l);
    }
    float den = 0.f, num = 0.f;
#pragma unroll
    for (int h = 0; h < H_; ++h) {
        float w = expf(lg[h] - mx);
        den += w;
        num = fmaf(so[((size_t)b * HS_ + pos[h]) * D_ + d], w, num);
    }
    out[((size_t)b * S_ + s) * D_ + d] = num / den;
}

// ---------------------------------------------------------------------------
extern "C" void kernel_launch(void* const* d_in, const int* in_sizes, int n_in,
                              void* d_out, int out_size, void* d_ws, size_t ws_size,
                              hipStream_t stream) {
    (void)in_sizes; (void)n_in; (void)out_size; (void)ws_size;
    const float* qk  = (const float*)d_in[0];
    const float* v   = (const float*)d_in[1];
    const float* rot = (const float*)d_in[2];
    float* out = (float*)d_out;

    char* ws = (char*)d_ws;
    const size_t n_tok = (size_t)B_ * HS_;           // 524288
    unsigned* bucket   = (unsigned*)ws; ws += n_tok * 4;
    unsigned* sorted_s = (unsigned*)ws; ws += n_tok * 4;
    unsigned* inv      = (unsigned*)ws; ws += n_tok * 4;
    float*    slog     = (float*)ws;    ws += n_tok * 4;
    float*    so       = (float*)ws;    // n_tok * 64 floats (128 MiB)

    k_hash<<<dim3(B_ * H_ * (S_ / 64)), dim3(256), 0, stream>>>(qk, rot, bucket);

    size_t lds2 = (size_t)(64 * 256 + 256) * sizeof(unsigned);
    k_sort<<<dim3(B_ * H_), dim3(256), lds2, stream>>>(bucket, sorted_s, inv);

    size_t lds3 = (size_t)(3 * 128 * 68 + 64 * 132) * sizeof(float)
                + 128 * sizeof(int) + 128 * sizeof(float);
    k_attn<<<dim3(B_ * C_), dim3(256), lds3, stream>>>(qk, v, sorted_s, so, slog);

    k_comb<<<dim3(B_ * S_ / 4), dim3(256), 0, stream>>>(so, slog, inv, out);
}